// GTConv_65472481460783
// MI455X (gfx1250) — compile-verified
//
#include <hip/hip_runtime.h>
#include <hip/hip_bf16.h>

// GTConv on MI455X (gfx1250).
//
// Output tuple (concatenated in d_out, per-doc int32 for integer tensors):
//   [0 .. 2*R*E)        total_edge_index (int32): out[s][r*E+e] = edge_index[r][s][e]
//   [2*R*E .. +C*R*E)   vals (f32): vals[c][r*E+e] = softmax(W,axis=0)[r][c] * ev[r][e]
//
// Roofline: ~576 MB total traffic @ 23.3 TB/s ~= 25 us. Pure streaming; the
// broadcast multiply is mapped onto V_WMMA_F32_16X16X4_F32 with a selector
// A-matrix so every one of the 256 D entries is a distinct useful product
// (4 channels x 64 edge values per WMMA). B load = one coalesced 8B/lane;
// D store = two coalesced 16B/lane.
//
// Pipeline: loop control is forced into SGPRs via readfirstlane so every
// branch is scalar (EXEC provably all-1s at the WMMA, as the ISA requires),
// and the next tile's B load is UNCONDITIONAL (tail handled by clamping the
// address with a scalar select), giving a constant 2-loads-in-flight
// invariant so the compiler can wait loadcnt<=1 and overlap load with
// WMMA + stores.

typedef __attribute__((ext_vector_type(2))) float v2f;
typedef __attribute__((ext_vector_type(8))) float v8f;

#define R_REL 8
#define C_CH  4

// ---------------- Kernel 1: index transpose (segment-wise uint4 memcpy) -----
// in:  [R][2][E] int32   out: [2][R*E] int32   (each (r,s) segment contiguous)
__global__ void gt_idx_copy(const uint4* __restrict__ in,
                            uint4* __restrict__ out, int e4 /* = E/4 */) {
    const int seg = blockIdx.y;            // seg = r*2 + s
    const int r = seg >> 1, s = seg & 1;
    const int i = blockIdx.x * blockDim.x + threadIdx.x;
    if (i < e4) {
        out[(size_t)(s * R_REL + r) * e4 + i] = in[(size_t)seg * e4 + i];
    }
}

// ---------------- Kernel 2: vals via WMMA outer product ---------------------
// Per relation r (blockIdx.y): D[m,n] = A[m,k]*B[k,n] with
//   A[m,k] = filt[r][m>>2] * (k == (m&3))      (built once per wave)
//   B[k,n] = ev[r][64*t + 4n + k]              (one 8B load per lane)
// => D[m,n] = filt[r][c] * ev[j],  c = m>>2, j = 64*t + 4n + (m&3).
__global__ void gt_vals_wmma(const float* __restrict__ weight,
                             const float* __restrict__ ev,
                             float* __restrict__ out, int E) {
    const int r    = blockIdx.y;
    const int lane = threadIdx.x & 31;
    const int m    = lane & 15;
    const int cA   = m >> 2;               // channel owned by this A-row
    const int k0   = m & 3;
    const int kx   = (lane < 16) ? 0 : 2;  // K held in VGPR0 for this lane half

    // softmax over relations (axis 0) for column cA; 32 floats, L2-resident.
    float wmax = -3.0e38f;
    #pragma unroll
    for (int j = 0; j < R_REL; ++j) wmax = fmaxf(wmax, weight[j * C_CH + cA]);
    float wsum = 0.0f;
    #pragma unroll
    for (int j = 0; j < R_REL; ++j) wsum += __expf(weight[j * C_CH + cA] - wmax);
    const float filt = __expf(weight[r * C_CH + cA] - wmax) / wsum;

    // A-matrix (16x4 f32, 2 VGPRs): VGPR0 holds K=kx, VGPR1 holds K=kx+1.
    v2f a;
    a.x = (k0 == kx)     ? filt : 0.0f;
    a.y = (k0 == kx + 1) ? filt : 0.0f;

    // Store mapping: D VGPRs 0..3 -> channel (lane<16 ? 0 : 2), 4..7 -> +1,
    // column j = 4*(lane&15) + k0 with k0 consecutive across the VGPR quad.
    const size_t CH   = (size_t)R_REL * (size_t)E;   // channel stride in vals
    const int    n    = lane & 15;
    const int    cLo  = (lane < 16) ? 0 : 2;
    const size_t rowB = (size_t)r * (size_t)E + (size_t)(4 * n);
    float* out0 = out + (size_t)cLo * CH + rowB;
    float* out1 = out + (size_t)(cLo + 1) * CH + rowB;

    const float* evp = ev + (size_t)r * (size_t)E + (4 * n + kx); // per-lane base

    // Scalarized (SGPR) loop control: wave id via readfirstlane so all loop
    // conditions compile to scalar branches (no EXEC manipulation near WMMA).
    const int wavesPerBlock = blockDim.x >> 5;
    const int waveInBlock   = __builtin_amdgcn_readfirstlane((int)(threadIdx.x >> 5));
    const int gw            = blockIdx.x * wavesPerBlock + waveInBlock;
    const int wavesTotal    = gridDim.x * wavesPerBlock;
    const int nTiles        = E >> 6;                 // 64 values per WMMA tile
    const int pfDist        = wavesTotal << 6;        // ~1 MB lookahead

    int t = gw;                                       // SGPR loop counter
    if (t < nTiles) {                                 // scalar branch
        v2f b = *(const v2f*)(evp + ((size_t)t << 6));  // prologue load
        for (;;) {
            const int tn = t + wavesTotal;
            const int tl = (tn < nTiles) ? tn : t;    // scalar clamp (s_cselect)
            // Unconditional next-tile load: keeps exactly 2 loads in flight so
            // the WMMA below only needs loadcnt<=1 (loads retire in order).
            v2f bn = *(const v2f*)(evp + ((size_t)tl << 6));
            __builtin_prefetch(evp + ((size_t)tl << 6) + pfDist, 0, 1); // global_prefetch_b8
            v8f c = {};
            // D = A x B + 0   (v_wmma_f32_16x16x4_f32)
            v8f d = __builtin_amdgcn_wmma_f32_16x16x4_f32(
                        /*neg_a=*/false, a, /*neg_b=*/false, b,
                        /*c_mod=*/(short)0, c, /*reuse_a=*/false, /*reuse_b=*/false);
            const size_t base = (size_t)t << 6;
            float4 q0 = make_float4(d[0], d[1], d[2], d[3]); // channel cLo
            float4 q1 = make_float4(d[4], d[5], d[6], d[7]); // channel cLo+1
            *(float4*)(out0 + base) = q0;                    // coalesced 16B/lane
            *(float4*)(out1 + base) = q1;
            b = bn;
            t = tn;
            if (t >= nTiles) break;                   // scalar branch
        }
    }
}

// ---------------------------------------------------------------------------
extern "C" void kernel_launch(void* const* d_in, const int* in_sizes, int n_in,
                              void* d_out, int out_size, void* d_ws, size_t ws_size,
                              hipStream_t stream) {
    const float* weight = (const float*)d_in[0];   // [8,4] f32
    const int*   eidx   = (const int*)d_in[1];     // [8,2,E] int32 (per bridge doc)
    const float* ev     = (const float*)d_in[2];   // [8,E] f32

    const int E = in_sizes[2] / R_REL;             // 2,000,000 (divisible by 64)
    const long long idxElems = 2LL * R_REL * E;    // 32M int32 slots

    int*   out_idx  = (int*)d_out;
    float* out_vals = (float*)d_out + idxElems;

    // Kernel 1: 16 segments x E/4 uint4 copies
    const int e4 = E >> 2;
    dim3 b1(256), g1((unsigned)((e4 + 255) / 256), 2 * R_REL);
    gt_idx_copy<<<g1, b1, 0, stream>>>((const uint4*)eidx, (uint4*)out_idx, e4);

    // Kernel 2: 8 relations x 512 blocks x 8 waves, grid-stride over 64-wide tiles
    dim3 b2(256), g2(512, R_REL);
    gt_vals_wmma<<<g2, b2, 0, stream>>>(weight, ev, out_vals, E);
}